// BinarizeLayer_59794534695076
// MI455X (gfx1250) — compile-verified
//
#include <hip/hip_runtime.h>
#include <stdint.h>

// Min s-t cut of a chain == 2-state Viterbi == scan over 2x2 (min,+) tropical
// matrices. 3-phase block-scan decomposition; labels via argmin(alpha+beta).
// BW-bound: ~100MB traffic => ~4.3us floor @ 23.3 TB/s. Staging uses CDNA5
// async global->LDS copies (ASYNCcnt) with an XOR swizzle on LDS destinations
// so per-thread ds_load_b128 reads are bank-conflict-free.

#define LAM  0.75f
#define TINF 1e30f

constexpr int TPB   = 256;          // 8 waves (wave32)
constexpr int EPT   = 16;           // elements per thread
constexpr int CHUNK = TPB * EPT;    // 4096 floats per workgroup
constexpr int NTOT  = 8388608;
constexpr int NBLK  = NTOT / CHUNK; // 2048 blocks
constexpr int PER   = NBLK / TPB;   // 8 blocks per thread in phase 2

typedef int v4i __attribute__((ext_vector_type(4)));

struct __align__(16) TM { float m00, m01, m10, m11; };

__device__ __forceinline__ TM tm_ident() { TM r = {0.f, TINF, TINF, 0.f}; return r; }
__device__ __forceinline__ TM tm_T(const TM& a) { TM r = {a.m00, a.m10, a.m01, a.m11}; return r; }

// R = L (later) ⊗ E (earlier) in (min,+); renormalized.
__device__ __forceinline__ TM tm_compose(const TM& L, const TM& E) {
  TM r;
  r.m00 = fminf(L.m00 + E.m00, L.m01 + E.m10);
  r.m01 = fminf(L.m00 + E.m01, L.m01 + E.m11);
  r.m10 = fminf(L.m10 + E.m00, L.m11 + E.m10);
  r.m11 = fminf(L.m10 + E.m01, L.m11 + E.m11);
  float m = fminf(fminf(r.m00, r.m01), fminf(r.m10, r.m11));
  r.m00 -= m; r.m01 -= m; r.m10 -= m; r.m11 -= m;
  return r;
}

// Fold one element matrix [[c0,c0+L],[c1+L,c1]] on the left of accumulator A.
__device__ __forceinline__ TM tm_fold(const TM& A, float c0, float c1) {
  TM r;
  r.m00 = c0 + fminf(A.m00, A.m10 + LAM);
  r.m01 = c0 + fminf(A.m01, A.m11 + LAM);
  r.m10 = c1 + fminf(A.m10, A.m00 + LAM);
  r.m11 = c1 + fminf(A.m11, A.m01 + LAM);
  return r;
}

__device__ __forceinline__ void apply_dir(const TM& q, float& x0, float& x1) {
  float n0 = fminf(q.m00 + x0, q.m01 + x1);
  float n1 = fminf(q.m10 + x0, q.m11 + x1);
  float m = fminf(n0, n1);
  x0 = n0 - m; x1 = n1 - m;
}
__device__ __forceinline__ void apply_T(const TM& q, float& x0, float& x1) {
  float n0 = fminf(q.m00 + x0, q.m10 + x1);
  float n1 = fminf(q.m01 + x0, q.m11 + x1);
  float m = fminf(n0, n1);
  x0 = n0 - m; x1 = n1 - m;
}

// ---------------- async global -> LDS staging (CDNA5) ----------------------

// swizzle 16B-chunk slot so reader thread t's chunks {4t..4t+3} land on
// distinct bank-quads across each half-wave.
__device__ __forceinline__ uint32_t swz(uint32_t c) { return c ^ ((c >> 4) & 3u); }

__device__ __forceinline__ void async_copy_b128(const float* src, float* smem_base,
                                                uint32_t dst_byte) {
  char* dst = (char*)smem_base + dst_byte;
#if __has_builtin(__builtin_amdgcn_global_load_async_to_lds_b128)
  __builtin_amdgcn_global_load_async_to_lds_b128(
      (__attribute__((address_space(1))) v4i*)(void*)src,
      (__attribute__((address_space(3))) v4i*)dst, 0, 0);
#else
  uint32_t ldsoff = (uint32_t)(size_t)(__attribute__((address_space(3))) char*)dst;
  asm volatile("global_load_async_to_lds_b128 %0, %1, off"
               :: "v"(ldsoff), "v"(src) : "memory");
#endif
}

__device__ __forceinline__ void async_wait_all() {
#if __has_builtin(__builtin_amdgcn_s_wait_asynccnt)
  __builtin_amdgcn_s_wait_asynccnt(0);
#else
  asm volatile("s_wait_asynccnt 0" ::: "memory");
#endif
}

// Stage CHUNK floats (16KB) into LDS; lane i copies 16B => fully coalesced.
__device__ __forceinline__ void stage(const float* __restrict__ gsrc,
                                      float* smem, int t) {
#pragma unroll
  for (int k = 0; k < 4; ++k) {
    uint32_t c = (uint32_t)(k * TPB + t);          // 16B chunk index 0..1023
    async_copy_b128(gsrc + c * 4, smem, swz(c) * 16u);
  }
  async_wait_all();
  __syncthreads();
}

__device__ __forceinline__ float4 read_chunk(const float* smem, uint32_t chunk) {
  return *(const float4*)(smem + swz(chunk) * 4);
}

__device__ __forceinline__ TM build_thread_mat(const float* e) {
  float c0 = e[0], c1 = 1.0f - e[0];
  TM A = {c0, c0 + LAM, c1 + LAM, c1};
#pragma unroll
  for (int j = 1; j < EPT; ++j) A = tm_fold(A, e[j], 1.0f - e[j]);
  float m = fminf(fminf(A.m00, A.m01), fminf(A.m10, A.m11));
  A.m00 -= m; A.m01 -= m; A.m10 -= m; A.m11 -= m;
  return A;
}

// Hillis-Steele inclusive scan (non-commutative, order-preserving) over TPB
// tropical matrices in LDS double buffer. Returns buffer holding the result.
__device__ __forceinline__ int hs_scan(TM (&sc)[2][TPB], int t) {
  int src = 0;
#pragma unroll
  for (int d = 1; d < TPB; d <<= 1) {
    TM v = sc[src][t];
    if (t >= d) v = tm_compose(v, sc[src][t - d]);
    sc[src ^ 1][t] = v;
    __syncthreads();
    src ^= 1;
  }
  return src;
}

// --------------------------- phase 1 ---------------------------------------
__global__ void __launch_bounds__(TPB)
dp_phase1(const float* __restrict__ p, TM* __restrict__ Q) {
  __shared__ __align__(16) float smem[CHUNK];
  __shared__ TM red[TPB];
  const int t = threadIdx.x;
  stage(p + (size_t)blockIdx.x * CHUNK, smem, t);

  float4 v[4];
#pragma unroll
  for (int k = 0; k < 4; ++k) v[k] = read_chunk(smem, (uint32_t)(4 * t + k));
  float e[EPT];
#pragma unroll
  for (int k = 0; k < 4; ++k) {
    e[4 * k + 0] = v[k].x; e[4 * k + 1] = v[k].y;
    e[4 * k + 2] = v[k].z; e[4 * k + 3] = v[k].w;
  }
  red[t] = build_thread_mat(e);
  __syncthreads();
  // order-preserving tree reduction: later ⊗ earlier
#pragma unroll
  for (int s = 1; s < TPB; s <<= 1) {
    bool act = (t & (2 * s - 1)) == 0;
    TM merged;
    if (act) merged = tm_compose(red[t + s], red[t]);
    __syncthreads();
    if (act) red[t] = merged;
    __syncthreads();
  }
  if (t == 0) Q[blockIdx.x] = red[0];
}

// --------------------------- phase 2 ---------------------------------------
__global__ void __launch_bounds__(TPB)
dp_phase2(const TM* __restrict__ Q, float2* __restrict__ aIn,
          float2* __restrict__ bIn) {
  __shared__ TM sc[2][TPB];
  const int t = threadIdx.x;
  TM q[PER];
  TM A = tm_ident();
#pragma unroll
  for (int j = 0; j < PER; ++j) {
    q[j] = Q[t * PER + j];
    A = (j == 0) ? q[0] : tm_compose(q[j], A);
  }
  // ---- forward scan: alpha entering each block
  sc[0][t] = A;
  __syncthreads();
  int s1 = hs_scan(sc, t);
  float a0 = 0.f, a1 = 0.f;
  if (t > 0) {
    TM ex = sc[s1][t - 1];
    a0 = fminf(ex.m00, ex.m01); a1 = fminf(ex.m10, ex.m11); // ex ⊗ [0,0]
    float m = fminf(a0, a1); a0 -= m; a1 -= m;
  }
#pragma unroll
  for (int j = 0; j < PER; ++j) {
    aIn[t * PER + j] = make_float2(a0, a1);
    apply_dir(q[j], a0, a1);
  }
  __syncthreads();
  // ---- backward scan: beta at right edge of each block.
  // Backward composite of a segment == transpose of its forward composite.
  sc[0][TPB - 1 - t] = tm_T(A);
  __syncthreads();
  int s2 = hs_scan(sc, t);
  int u = TPB - 1 - t;
  float b0 = 0.f, b1 = 0.f;
  if (u > 0) {
    TM ex = sc[s2][u - 1];
    b0 = fminf(ex.m00, ex.m01); b1 = fminf(ex.m10, ex.m11);
    float m = fminf(b0, b1); b0 -= m; b1 -= m;
  }
#pragma unroll
  for (int j = PER - 1; j >= 0; --j) {
    bIn[t * PER + j] = make_float2(b0, b1);
    apply_T(q[j], b0, b1);
  }
}

// --------------------------- phase 3 ---------------------------------------
__global__ void __launch_bounds__(TPB)
dp_phase3(const float* __restrict__ p, const float2* __restrict__ aInB,
          const float2* __restrict__ bInB, int* __restrict__ out) {
  __shared__ __align__(16) float smem[CHUNK];
  __shared__ TM sc[2][TPB];
  const int t = threadIdx.x;
  const int b = blockIdx.x;
  stage(p + (size_t)b * CHUNK, smem, t);

  float4 v[4];
#pragma unroll
  for (int k = 0; k < 4; ++k) v[k] = read_chunk(smem, (uint32_t)(4 * t + k));
  float e[EPT];
#pragma unroll
  for (int k = 0; k < 4; ++k) {
    e[4 * k + 0] = v[k].x; e[4 * k + 1] = v[k].y;
    e[4 * k + 2] = v[k].z; e[4 * k + 3] = v[k].w;
  }
  TM A = build_thread_mat(e);

  // forward intra-block scan -> alpha entering this thread's 16 elements
  sc[0][t] = A;
  __syncthreads();
  int s1 = hs_scan(sc, t);
  float2 ain = aInB[b];
  float a0 = ain.x, a1 = ain.y;
  if (t > 0) apply_dir(sc[s1][t - 1], a0, a1);
  __syncthreads();

  // backward intra-block scan -> beta at right edge of this thread's elements
  sc[0][TPB - 1 - t] = tm_T(A);
  __syncthreads();
  int s2 = hs_scan(sc, t);
  int u = TPB - 1 - t;
  float2 bin = bInB[b];
  float b0 = bin.x, b1 = bin.y;
  if (u > 0) apply_dir(sc[s2][u - 1], b0, b1);

  // per-element forward sweep (store alphas in registers)
  float al0[EPT], al1[EPT];
#pragma unroll
  for (int j = 0; j < EPT; ++j) {
    float c0 = e[j], c1 = 1.0f - e[j];
    float n0 = c0 + fminf(a0, a1 + LAM);
    float n1 = c1 + fminf(a1, a0 + LAM);
    float m = fminf(n0, n1);
    a0 = n0 - m; a1 = n1 - m;
    al0[j] = a0; al1[j] = a1;
  }
  // backward sweep + labels: x_j = argmin(alpha_j + beta_j)
  int lab[EPT];
#pragma unroll
  for (int j = EPT - 1; j >= 0; --j) {
    lab[j] = (al1[j] + b1 < al0[j] + b0) ? 1 : 0;
    float c0 = e[j], c1 = 1.0f - e[j];
    float n0 = fminf(c0 + b0, c1 + LAM + b1);
    float n1 = fminf(c0 + LAM + b0, c1 + b1);
    float m = fminf(n0, n1);
    b0 = n0 - m; b1 = n1 - m;
  }
  int4* o = (int4*)(out + (size_t)b * CHUNK + (size_t)t * EPT);
  o[0] = make_int4(lab[0],  lab[1],  lab[2],  lab[3]);
  o[1] = make_int4(lab[4],  lab[5],  lab[6],  lab[7]);
  o[2] = make_int4(lab[8],  lab[9],  lab[10], lab[11]);
  o[3] = make_int4(lab[12], lab[13], lab[14], lab[15]);
}

// ---------------------------------------------------------------------------
extern "C" void kernel_launch(void* const* d_in, const int* in_sizes, int n_in,
                              void* d_out, int out_size, void* d_ws, size_t ws_size,
                              hipStream_t stream) {
  (void)n_in; (void)out_size; (void)ws_size;
  const float* p = (const float*)d_in[0];
  int* out = (int*)d_out;
  const int n = in_sizes[0];       // 8388608
  const int nblk = n / CHUNK;      // 2048

  TM*     Q   = (TM*)d_ws;                                              // 32 KB
  float2* aIn = (float2*)((char*)d_ws + (size_t)nblk * sizeof(TM));     // 16 KB
  float2* bIn = (float2*)((char*)aIn + (size_t)nblk * sizeof(float2));  // 16 KB

  dp_phase1<<<nblk, TPB, 0, stream>>>(p, Q);
  dp_phase2<<<1,    TPB, 0, stream>>>(Q, aIn, bIn);
  dp_phase3<<<nblk, TPB, 0, stream>>>(p, aIn, bIn, out);
}